// GraphNet_43937515438735
// MI455X (gfx1250) — compile-verified
//
#include <hip/hip_runtime.h>

// GraphNetwork (jraph-style) for MI455X / gfx1250.
//
// Perf model: ~26 GFLOP vs ~1.3 GB HBM traffic -> memory bound at 23.3 TB/s
// only if compute runs >≈460 TFLOP/s, hence V_WMMA_F32_16X16X4_F32 (fp32-exact,
// matrix-unit rate). Edge/node MLPs are 16-wide so a 16x16 WMMA tile (one
// wave32 per 16 rows) fits exactly. Accumulators / node features are
// L2-resident (192MB L2); edge features stream in-place.
//
// v2: templated feature widths (full unroll, static chunk routing),
//     rcp-based SiLU (no IEEE divide), 32-bit unsigned offsets everywhere
//     (saddr+voffset addressing instead of 64-bit VALU address math).

typedef __attribute__((ext_vector_type(2))) float v2f;
typedef __attribute__((ext_vector_type(8))) float v8f;

#define WAVES_PER_BLOCK 8
#define BLOCK_THREADS (WAVES_PER_BLOCK * 32)
#define KPAD_MAX 56   // max padded input dim across all MLPs

struct SharedMLP {
    float W1[KPAD_MAX * 16];
    float b1[16];
    float W2[16 * 16];
    float b2[16];
    float H[WAVES_PER_BLOCK * 16 * 16];  // layer-1 activations, per wave
};

__device__ __forceinline__ float silu_f(float x) {
    // x * rcp(1 + exp(-x)) : v_exp_f32 + v_rcp_f32 instead of IEEE divide
    return x * __builtin_amdgcn_rcpf(1.0f + __expf(-x));
}

__device__ __forceinline__ void load_weights(SharedMLP& sm,
                                             const float* __restrict__ W1,
                                             const float* __restrict__ b1,
                                             const float* __restrict__ W2,
                                             const float* __restrict__ b2,
                                             int Kin, int Kpad) {
    for (int i = threadIdx.x; i < Kpad * 16; i += blockDim.x) {
        int row = i >> 4;
        sm.W1[i] = (row < Kin) ? W1[i] : 0.0f;   // zero-pad K to multiple of 4
    }
    for (int i = threadIdx.x; i < 256; i += blockDim.x) sm.W2[i] = W2[i];
    if (threadIdx.x < 16) {
        sm.b1[threadIdx.x] = b1[threadIdx.x];
        sm.b2[threadIdx.x] = b2[threadIdx.x];
    }
    __syncthreads();
}

// ---------------- Edge update kernel ----------------
// x = concat(edge_src[e] (EW), node_src[senders[e]] (NW), node_src[receivers[e]] (NW), g (GW))
// y = silu(x@W1+b1) @ W2 + b2 -> edge_out[e]; scatter-add into sent/recv; sum into edge_sum
template<int EW, int NW, int GW>
__global__ void __launch_bounds__(BLOCK_THREADS)
gn_edge_kernel(const float* __restrict__ edge_src,
               const float* __restrict__ node_src,
               const float* __restrict__ gvec,
               const int* __restrict__ senders, const int* __restrict__ receivers,
               const float* __restrict__ W1, const float* __restrict__ b1,
               const float* __restrict__ W2, const float* __restrict__ b2,
               int numTiles,
               float* __restrict__ edge_out,
               float* __restrict__ sent_acc, float* __restrict__ recv_acc,
               float* __restrict__ edge_sum)
{
    constexpr int Kin   = EW + 2 * NW + GW;
    constexpr int Kpad  = (Kin + 3) & ~3;
    constexpr int NCHK  = Kpad >> 2;
    constexpr int sBase = EW;
    constexpr int rBase = EW + NW;
    constexpr int gBase = EW + 2 * NW;

    __shared__ SharedMLP sm;
    load_weights(sm, W1, b1, W2, b2, Kin, Kpad);

    const unsigned lane = threadIdx.x & 31u;
    const int waveInBlock = threadIdx.x >> 5;
    const int wave  = blockIdx.x * WAVES_PER_BLOCK + waveInBlock;
    const int nWav  = gridDim.x * WAVES_PER_BLOCK;
    const unsigned m    = lane & 15u;   // A-matrix row (M = lane % 16)
    const unsigned half = lane >> 4;    // 0 -> K+0/K+1 ; 1 -> K+2/K+3 (16x4 f32 A layout)
    const unsigned n    = lane & 15u;   // B/C/D column (N = lane % 16)
    float* Hs = &sm.H[waveInBlock * 256];

    float psum = 0.0f;  // partial of edge_sum[n]

    for (int tile = wave; tile < numTiles; tile += nWav) {
        const unsigned ebase = (unsigned)tile << 4;
        const unsigned e = ebase + m;
        const unsigned s = (unsigned)senders[e];
        const unsigned r = (unsigned)receivers[e];

        v8f acc = {0.f, 0.f, 0.f, 0.f, 0.f, 0.f, 0.f, 0.f};
#pragma unroll
        for (int c = 0; c < NCHK; ++c) {
            const int k0 = c << 2;
            float4 x;
            if (k0 < sBase) {
                x = *(const float4*)(edge_src + e * (unsigned)EW + (unsigned)k0);
            } else if (k0 < rBase) {
                x = *(const float4*)(node_src + s * (unsigned)NW + (unsigned)(k0 - sBase));
            } else if (k0 < gBase) {
                x = *(const float4*)(node_src + r * (unsigned)NW + (unsigned)(k0 - rBase));
            } else {
                const int gi = k0 - gBase;
                x.x = (gi + 0 < GW) ? gvec[gi + 0] : 0.0f;
                x.y = (gi + 1 < GW) ? gvec[gi + 1] : 0.0f;
                x.z = (gi + 2 < GW) ? gvec[gi + 2] : 0.0f;
                x.w = (gi + 3 < GW) ? gvec[gi + 3] : 0.0f;
            }
            v2f a; a.x = half ? x.z : x.x; a.y = half ? x.w : x.y;
            const unsigned row = (unsigned)k0 + (half << 1);
            v2f b; b.x = sm.W1[row * 16 + n]; b.y = sm.W1[(row + 1) * 16 + n];
            acc = __builtin_amdgcn_wmma_f32_16x16x4_f32(false, a, false, b,
                                                        (short)0, acc, false, false);
        }
        // bias + SiLU, stage H (C-layout -> LDS) for layer-2 A fragments
#pragma unroll
        for (int j = 0; j < 8; ++j) {
            const unsigned mrow = (unsigned)j + (half << 3);
            Hs[mrow * 16 + n] = silu_f(acc[j] + sm.b1[n]);
        }
        v8f acc2 = {0.f, 0.f, 0.f, 0.f, 0.f, 0.f, 0.f, 0.f};
#pragma unroll
        for (int c = 0; c < 4; ++c) {
            const unsigned row = ((unsigned)c << 2) + (half << 1);
            v2f a; a.x = Hs[m * 16 + row]; a.y = Hs[m * 16 + row + 1];
            v2f b; b.x = sm.W2[row * 16 + n]; b.y = sm.W2[(row + 1) * 16 + n];
            acc2 = __builtin_amdgcn_wmma_f32_16x16x4_f32(false, a, false, b,
                                                         (short)0, acc2, false, false);
        }
#pragma unroll
        for (int j = 0; j < 8; ++j) {
            const unsigned mrow = (unsigned)j + (half << 3);
            const unsigned eo = ebase + mrow;
            const float val = acc2[j] + sm.b2[n];
            edge_out[eo * 16u + n] = val;
            atomicAdd(sent_acc + (unsigned)senders[eo] * 16u + n, val);
            atomicAdd(recv_acc + (unsigned)receivers[eo] * 16u + n, val);
            psum += val;
        }
    }
    atomicAdd(edge_sum + n, psum);
}

// ---------------- Node update kernel ----------------
// x = concat(node_src[v] (NW0), sent[v] (16), recv[v] (16), g (GW))
template<int NW0, int GW>
__global__ void __launch_bounds__(BLOCK_THREADS)
gn_node_kernel(const float* __restrict__ node_src,
               const float* __restrict__ sent_acc, const float* __restrict__ recv_acc,
               const float* __restrict__ gvec,
               const float* __restrict__ W1, const float* __restrict__ b1,
               const float* __restrict__ W2, const float* __restrict__ b2,
               int numTiles,
               float* __restrict__ node_out, float* __restrict__ node_sum)
{
    constexpr int Kin   = NW0 + 32 + GW;
    constexpr int Kpad  = (Kin + 3) & ~3;
    constexpr int NCHK  = Kpad >> 2;
    constexpr int sBase = NW0;
    constexpr int rBase = NW0 + 16;
    constexpr int gBase = NW0 + 32;

    __shared__ SharedMLP sm;
    load_weights(sm, W1, b1, W2, b2, Kin, Kpad);

    const unsigned lane = threadIdx.x & 31u;
    const int waveInBlock = threadIdx.x >> 5;
    const int wave  = blockIdx.x * WAVES_PER_BLOCK + waveInBlock;
    const int nWav  = gridDim.x * WAVES_PER_BLOCK;
    const unsigned m    = lane & 15u;
    const unsigned half = lane >> 4;
    const unsigned n    = lane & 15u;
    float* Hs = &sm.H[waveInBlock * 256];

    float psum = 0.0f;  // partial of node_sum[n]

    for (int tile = wave; tile < numTiles; tile += nWav) {
        const unsigned nbase = (unsigned)tile << 4;
        const unsigned rowIdx = nbase + m;

        v8f acc = {0.f, 0.f, 0.f, 0.f, 0.f, 0.f, 0.f, 0.f};
#pragma unroll
        for (int c = 0; c < NCHK; ++c) {
            const int k0 = c << 2;
            float4 x;
            if (k0 < sBase) {
                x = *(const float4*)(node_src + rowIdx * (unsigned)NW0 + (unsigned)k0);
            } else if (k0 < rBase) {
                x = *(const float4*)(sent_acc + rowIdx * 16u + (unsigned)(k0 - sBase));
            } else if (k0 < gBase) {
                x = *(const float4*)(recv_acc + rowIdx * 16u + (unsigned)(k0 - rBase));
            } else {
                const int gi = k0 - gBase;
                x.x = (gi + 0 < GW) ? gvec[gi + 0] : 0.0f;
                x.y = (gi + 1 < GW) ? gvec[gi + 1] : 0.0f;
                x.z = (gi + 2 < GW) ? gvec[gi + 2] : 0.0f;
                x.w = (gi + 3 < GW) ? gvec[gi + 3] : 0.0f;
            }
            v2f a; a.x = half ? x.z : x.x; a.y = half ? x.w : x.y;
            const unsigned row = (unsigned)k0 + (half << 1);
            v2f b; b.x = sm.W1[row * 16 + n]; b.y = sm.W1[(row + 1) * 16 + n];
            acc = __builtin_amdgcn_wmma_f32_16x16x4_f32(false, a, false, b,
                                                        (short)0, acc, false, false);
        }
#pragma unroll
        for (int j = 0; j < 8; ++j) {
            const unsigned mrow = (unsigned)j + (half << 3);
            Hs[mrow * 16 + n] = silu_f(acc[j] + sm.b1[n]);
        }
        v8f acc2 = {0.f, 0.f, 0.f, 0.f, 0.f, 0.f, 0.f, 0.f};
#pragma unroll
        for (int c = 0; c < 4; ++c) {
            const unsigned row = ((unsigned)c << 2) + (half << 1);
            v2f a; a.x = Hs[m * 16 + row]; a.y = Hs[m * 16 + row + 1];
            v2f b; b.x = sm.W2[row * 16 + n]; b.y = sm.W2[(row + 1) * 16 + n];
            acc2 = __builtin_amdgcn_wmma_f32_16x16x4_f32(false, a, false, b,
                                                         (short)0, acc2, false, false);
        }
#pragma unroll
        for (int j = 0; j < 8; ++j) {
            const unsigned mrow = (unsigned)j + (half << 3);
            const float val = acc2[j] + sm.b2[n];
            node_out[(nbase + mrow) * 16u + n] = val;
            psum += val;
        }
    }
    atomicAdd(node_sum + n, psum);
}

// ---------------- Global (graph-level) MLP ----------------
__global__ void gn_global_kernel(const float* __restrict__ node_sum,
                                 const float* __restrict__ edge_sum,
                                 const float* __restrict__ g_in, int gw,
                                 const float* __restrict__ W1, const float* __restrict__ b1,
                                 const float* __restrict__ W2, const float* __restrict__ b2,
                                 int gout, float* __restrict__ g_out,
                                 float* __restrict__ final_out)
{
    if (threadIdx.x != 0 || blockIdx.x != 0) return;
    const int Kin = 32 + gw;
    float x[40];
    for (int i = 0; i < 16; ++i) x[i]      = node_sum[i];
    for (int i = 0; i < 16; ++i) x[16 + i] = edge_sum[i];
    for (int i = 0; i < gw; ++i) x[32 + i] = g_in[i];
    float h[8];
    for (int j = 0; j < 8; ++j) {
        float a = b1[j];
        for (int k = 0; k < Kin; ++k) a += x[k] * W1[k * 8 + j];
        h[j] = silu_f(a);
    }
    for (int j = 0; j < gout; ++j) {
        float a = b2[j];
        for (int k = 0; k < 8; ++k) a += h[k] * W2[k * gout + j];
        g_out[j] = a;
        if (final_out && j == 0) final_out[0] = a;
    }
}

__global__ void zero_kernel(float* __restrict__ p, size_t n) {
    size_t i = (size_t)blockIdx.x * blockDim.x + threadIdx.x;
    const size_t stride = (size_t)gridDim.x * blockDim.x;
    for (; i < n; i += stride) p[i] = 0.0f;
}

// ---------------- Host orchestration ----------------
extern "C" void kernel_launch(void* const* d_in, const int* in_sizes, int n_in,
                              void* d_out, int out_size, void* d_ws, size_t ws_size,
                              hipStream_t stream)
{
    (void)n_in; (void)out_size; (void)ws_size;
    const int N = 200000, E = 2000000;

    const float* nodes0    = (const float*)d_in[0];
    const float* edges0    = (const float*)d_in[1];
    const int*   senders   = (const int*)d_in[2];
    const int*   receivers = (const int*)d_in[3];

    // Param dicts flatten as 3 sub-MLPs x (W0,b0,W1,b1). Detect key order
    // (sorted 'edge','global','node' vs insertion 'edge','node','global')
    // from the encoder W0 sizes: edge 25*16, global 33*8, node 41*16.
    int off_e = 0, off_g = 4, off_n = 8;
    for (int s = 0; s < 3; ++s) {
        const int w0 = in_sizes[4 + 4 * s];
        if      (w0 == 25 * 16) off_e = 4 * s;
        else if (w0 == 33 * 8)  off_g = 4 * s;
        else if (w0 == 41 * 16) off_n = 4 * s;
    }
    auto F = [&](int base, int off, int idx) { return (const float*)d_in[base + off + idx]; };

    // workspace layout
    char* ws = (char*)d_ws;
    size_t off = 0;
    auto wsAlloc = [&](size_t bytes) {
        void* p = ws + off;
        off = (off + bytes + 255) & ~(size_t)255;
        return p;
    };
    float* edge_feat = (float*)wsAlloc((size_t)E * 16 * sizeof(float)); // 128 MB
    float* node_feat = (float*)wsAlloc((size_t)N * 16 * sizeof(float));
    float* sent      = (float*)wsAlloc((size_t)N * 16 * sizeof(float));
    float* recv      = (float*)wsAlloc((size_t)N * 16 * sizeof(float));
    float* nsum      = (float*)wsAlloc(64);
    float* esum      = (float*)wsAlloc(64);
    float* g_a       = (float*)wsAlloc(64);
    float* g_b       = (float*)wsAlloc(64);

    auto zero_accums = [&]() {
        zero_kernel<<<256, 256, 0, stream>>>(sent, (size_t)N * 16);
        zero_kernel<<<256, 256, 0, stream>>>(recv, (size_t)N * 16);
        zero_kernel<<<1, 64, 0, stream>>>(nsum, 16);
        zero_kernel<<<1, 64, 0, stream>>>(esum, 16);
    };

    // one processor/decoder-shaped GN block (16/16/8 widths)
    auto run_proc_block = [&](const float* g_in, int pbase, int gout,
                              float* g_out, float* fin) {
        zero_accums();
        gn_edge_kernel<16, 16, 8><<<1024, BLOCK_THREADS, 0, stream>>>(
            edge_feat, node_feat, g_in, senders, receivers,
            F(pbase, off_e, 0), F(pbase, off_e, 1), F(pbase, off_e, 2), F(pbase, off_e, 3),
            E / 16, edge_feat, sent, recv, esum);
        gn_node_kernel<16, 8><<<512, BLOCK_THREADS, 0, stream>>>(
            node_feat, sent, recv, g_in,
            F(pbase, off_n, 0), F(pbase, off_n, 1), F(pbase, off_n, 2), F(pbase, off_n, 3),
            N / 16, node_feat, nsum);
        gn_global_kernel<<<1, 32, 0, stream>>>(
            nsum, esum, g_in, 8,
            F(pbase, off_g, 0), F(pbase, off_g, 1), F(pbase, off_g, 2), F(pbase, off_g, 3),
            gout, g_out, fin);
    };

    // encoder globals start at zero
    zero_kernel<<<1, 64, 0, stream>>>(g_a, 16);

    // ---- encoder (raw 8-dim features, g width 1) -> edge_feat/node_feat/g_b
    zero_accums();
    gn_edge_kernel<8, 8, 1><<<1024, BLOCK_THREADS, 0, stream>>>(
        edges0, nodes0, g_a, senders, receivers,
        F(4, off_e, 0), F(4, off_e, 1), F(4, off_e, 2), F(4, off_e, 3),
        E / 16, edge_feat, sent, recv, esum);
    gn_node_kernel<8, 1><<<512, BLOCK_THREADS, 0, stream>>>(
        nodes0, sent, recv, g_a,
        F(4, off_n, 0), F(4, off_n, 1), F(4, off_n, 2), F(4, off_n, 3),
        N / 16, node_feat, nsum);
    gn_global_kernel<<<1, 32, 0, stream>>>(
        nsum, esum, g_a, 1,
        F(4, off_g, 0), F(4, off_g, 1), F(4, off_g, 2), F(4, off_g, 3),
        8, g_b, nullptr);

    // ---- 3 processor steps (shared params, in-place features, g ping-pong)
    run_proc_block(g_b, 16, 8, g_a, nullptr);
    run_proc_block(g_a, 16, 8, g_b, nullptr);
    run_proc_block(g_b, 16, 8, g_a, nullptr);
    // ---- decoder: global MLP outputs 1 scalar -> d_out
    run_proc_block(g_a, 28, 1, g_b, (float*)d_out);
}